// Main_net_63599875719484
// MI455X (gfx1250) — compile-verified
//
#include <hip/hip_runtime.h>
#include <math.h>

// ---------------- problem constants (fixed by reference) ----------------
#define B_   4
#define H1_  256
#define W1_  448
#define H2_  128
#define W2_  224
#define H4_  64
#define W4_  112

// ---------------- workspace layout (float offsets) ----------------------
// raw (unscaled) upsampled flows, shape (2 dirs, B, 2, H, W)
#define RF_OFF    0u          // full res:   2*4*2*256*448 = 1,835,008
#define RH_OFF    1835008u    // half res:   2*4*2*128*224 =   458,752
#define M_OFF     2293760u    // metric full (2,B,H1,W1)   =   917,504
#define MH_OFF    3211264u    // metric half (2,B,H2,W2)   =   229,376
#define MQ_OFF    3440640u    // metric quarter (2,B,H4,W4)=    57,344
#define DEN12_1   3497984u    // (B,H1,W1) 458,752
#define DEN21_1   3956736u
#define DEN12_2   4415488u    // (B,H2,W2) 114,688
#define DEN21_2   4530176u
#define DEN12_3   4644864u    // (B,H4,W4) 28,672
#define DEN21_3   4673536u
#define DEN_OFF   DEN12_1
#define DEN_LEN   1204224u    // total den floats (contiguous)

// output layout (concatenated flat, floats)
#define OUT1_OFF  0u          // (4,70,256,448) = 32,112,640
#define OUT2_OFF  32112640u   // (4,128,128,224)= 14,680,064
#define OUT3_OFF  46792704u   // (4,192, 64,112)=  5,505,024
#define OUT_TOTAL 52297728u

// ---------------- zero fill (float4) ------------------------------------
__global__ void zero4_kernel(float4* __restrict__ p, long n4) {
  long i = (long)blockIdx.x * blockDim.x + threadIdx.x;
  long stride = (long)gridDim.x * blockDim.x;
  float4 z; z.x = 0.f; z.y = 0.f; z.z = 0.f; z.w = 0.f;
  for (; i < n4; i += stride) p[i] = z;
}

// ---------------- flow bilinear upsample with async LDS staging ---------
// One block per (dir, b, comp, output row). The two needed source rows
// (2 x 112 floats) are pulled into LDS with the CDNA5 async global->LDS
// path, waited on with s_wait_asynccnt, then every lane interpolates one
// output pixel from LDS. Jax half-pixel convention with edge clamp.
__global__ void upflow_kernel(const float* __restrict__ flow12,
                              const float* __restrict__ flow21,
                              float* __restrict__ out, int B, int Ho, int Wo) {
  __shared__ float srow[2 * W4_];   // 224 floats
  int t = threadIdx.x;
  int r = blockIdx.x;               // ((f*B + b)*2 + c)*Ho + y
  int y = r % Ho; r /= Ho;
  int c = r % 2;  r /= 2;
  int b = r % B;  int f = r / B;

  const float* fp = f ? flow21 : flow12;
  const float* mp = fp + (size_t)(b * 2 + c) * (H4_ * W4_);  // map base

  float iyf = (y + 0.5f) * ((float)H4_ / (float)Ho) - 0.5f;
  int   iy0 = (int)floorf(iyf);
  float wy  = iyf - (float)iy0;
  int r0 = min(max(iy0, 0), H4_ - 1);
  int r1 = min(max(iy0 + 1, 0), H4_ - 1);

  if (t < 2 * W4_) {
    int row = (t < W4_) ? r0 : r1;
    int col = (t < W4_) ? t  : (t - W4_);
    unsigned vo   = (unsigned)((row * W4_ + col) * 4);           // global byte off
    unsigned ldsa = (unsigned)(unsigned long long)(&srow[t]);    // LDS byte addr
    unsigned long long sbase = (unsigned long long)(const void*)mp;
    asm volatile("global_load_async_to_lds_b32 %0, %1, %2"
                 :: "v"(ldsa), "v"(vo), "s"(sbase) : "memory");
  }
  asm volatile("s_wait_asynccnt 0" ::: "memory");
  __syncthreads();

  if (t < Wo) {
    float ixf = (t + 0.5f) * ((float)W4_ / (float)Wo) - 0.5f;
    int   ix0 = (int)floorf(ixf);
    float wx  = ixf - (float)ix0;
    int cL = min(max(ix0, 0), W4_ - 1);
    int cR = min(max(ix0 + 1, 0), W4_ - 1);
    float top = srow[cL] * (1.f - wx) + srow[cR] * wx;
    float bot = srow[W4_ + cL] * (1.f - wx) + srow[W4_ + cR] * wx;
    float v = top * (1.f - wy) + bot * wy;
    out[((size_t)((f * B + b) * 2 + c) * Ho + y) * Wo + t] = v;
  }
}

// ---------------- photometric metric (backwarp + mean |diff|) ------------
// dir 0: m12 = mean_c |img1 - warp(img2, 20*rF12)| ; dir 1 symmetric.
__global__ void metric_kernel(const float* __restrict__ img1,
                              const float* __restrict__ img2,
                              const float* __restrict__ rF,
                              float* __restrict__ mOut,
                              int B, int H, int W) {
  int HW = H * W;
  int N = 2 * B * HW;
  int idx = blockIdx.x * blockDim.x + threadIdx.x;
  if (idx >= N) return;
  int d = idx / (B * HW);
  int rem = idx - d * (B * HW);
  int b = rem / HW; int pix = rem - b * HW;
  int y = pix / W, x = pix - y * W;

  const float* anchor = d ? img2 : img1;
  const float* warp   = d ? img1 : img2;
  const float* fl = rF + (size_t)d * (B * 2 * HW) + (size_t)(b * 2) * HW;

  float tx = (float)x + 20.0f * fl[pix];
  float ty = (float)y + 20.0f * fl[HW + pix];
  int x0 = (int)floorf(tx), y0 = (int)floorf(ty);
  float ax = tx - (float)x0, ay = ty - (float)y0;

  const float* wb = warp + (size_t)(b * 3) * HW;
  float s0 = 0.f, s1 = 0.f, s2 = 0.f;
#pragma unroll
  for (int k = 0; k < 4; k++) {
    int xi = x0 + (k & 1), yi = y0 + (k >> 1);
    float w = ((k & 1) ? ax : 1.f - ax) * ((k >> 1) ? ay : 1.f - ay);
    if (xi >= 0 && xi < W && yi >= 0 && yi < H) {
      int o = yi * W + xi;
      s0 += w * wb[o]; s1 += w * wb[HW + o]; s2 += w * wb[2 * HW + o];
    }
  }
  const float* ab = anchor + (size_t)(b * 3) * HW;
  mOut[idx] = (fabsf(ab[pix] - s0) + fabsf(ab[HW + pix] - s1) +
               fabsf(ab[2 * HW + pix] - s2)) * (1.0f / 3.0f);
}

// ---------------- antialiased metric downsample (jax resize semantics) --
// half: 4 taps/dim, weights (1,3,3,1)/8 starting at 2y-1
// quarter: 8 taps/dim, weights (1,3,5,7,7,5,3,1)/32 starting at 4y-2
// out-of-range taps dropped and result renormalized (zero-pad + renorm).
__global__ void down_kernel(const float* __restrict__ in, float* __restrict__ out,
                            int NB, int H, int W, int Ho, int Wo, int quarter) {
  int N = NB * Ho * Wo;
  int idx = blockIdx.x * blockDim.x + threadIdx.x;
  if (idx >= N) return;
  int n = idx / (Ho * Wo);
  int p = idx - n * (Ho * Wo);
  int y = p / Wo, x = p - y * Wo;

  float wt[8]; int taps, y0, x0;
  if (!quarter) {
    taps = 4; y0 = 2 * y - 1; x0 = 2 * x - 1;
    wt[0] = 1.f / 8; wt[1] = 3.f / 8; wt[2] = 3.f / 8; wt[3] = 1.f / 8;
  } else {
    taps = 8; y0 = 4 * y - 2; x0 = 4 * x - 2;
    wt[0] = 1.f / 32; wt[1] = 3.f / 32; wt[2] = 5.f / 32; wt[3] = 7.f / 32;
    wt[4] = 7.f / 32; wt[5] = 5.f / 32; wt[6] = 3.f / 32; wt[7] = 1.f / 32;
  }
  const float* base = in + (size_t)n * H * W;
  float swy = 0.f, swx = 0.f;
  for (int i = 0; i < taps; i++) { if (y0 + i >= 0 && y0 + i < H) swy += wt[i]; }
  for (int j = 0; j < taps; j++) { if (x0 + j >= 0 && x0 + j < W) swx += wt[j]; }
  float acc = 0.f;
  for (int i = 0; i < taps; i++) {
    int yy = y0 + i; if (yy < 0 || yy >= H) continue;
    float rowacc = 0.f;
    const float* rp = base + (size_t)yy * W;
    for (int j = 0; j < taps; j++) {
      int xx = x0 + j; if (xx < 0 || xx >= W) continue;
      rowacc += wt[j] * rp[xx];
    }
    acc += wt[i] * rowacc;
  }
  out[idx] = acc / (swy * swx);
}

// ---------------- forward softsplat scatter ------------------------------
// One thread per source pixel. Numerators scatter straight into the final
// concatenated output tensor (channel offset c0, total Ctot); denominator
// accumulated only when addDen=1 (img/feat pairs share flow+metric).
__global__ void splat_kernel(const float* __restrict__ src, int C,
                             const float* __restrict__ rflow, float fscale,
                             const float* __restrict__ metric,
                             const float* __restrict__ beta,
                             float* __restrict__ numBase, int c0, int Ctot,
                             float* __restrict__ den, int addDen,
                             int B, int Hs, int Ws) {
  int HW = Hs * Ws;
  int N = B * HW;
  int idx = blockIdx.x * blockDim.x + threadIdx.x;
  if (idx >= N) return;
  int b = idx / HW; int pix = idx - b * HW;
  int y = pix / Ws, x = pix - y * Ws;

  float fx = fscale * rflow[(size_t)(b * 2) * HW + pix];
  float fy = fscale * rflow[(size_t)(b * 2 + 1) * HW + pix];
  float tx = (float)x + fx, ty = (float)y + fy;
  int x0 = (int)floorf(tx), y0 = (int)floorf(ty);
  float ax = tx - (float)x0, ay = ty - (float)y0;

  float em = expf(beta[0] * metric[(size_t)b * HW + pix]);

  int ti[4]; float tw[4]; int nv = 0;
#pragma unroll
  for (int k = 0; k < 4; k++) {
    int xi = x0 + (k & 1), yi = y0 + (k >> 1);
    float w = ((k & 1) ? ax : 1.f - ax) * ((k >> 1) ? ay : 1.f - ay);
    if (xi >= 0 && xi < Ws && yi >= 0 && yi < Hs) { ti[nv] = yi * Ws + xi; tw[nv] = w; nv++; }
  }
  if (addDen) {
    float* dp = den + (size_t)b * HW;
    for (int k = 0; k < nv; k++) atomicAdd(&dp[ti[k]], em * tw[k]);
  }
  for (int c = 0; c < C; c++) {
    if (c + 1 < C)  // hint the HW*4-byte-strided channel walk
      __builtin_prefetch(src + (size_t)(b * C + c + 1) * HW + pix, 0, 0);
    float v = src[(size_t)(b * C + c) * HW + pix] * em;
    float* oc = numBase + (size_t)(b * Ctot + c0 + c) * HW;
    for (int k = 0; k < nv; k++) atomicAdd(&oc[ti[k]], v * tw[k]);
  }
}

// ---------------- normalize: out = num / (den==0 ? 1 : den) --------------
__global__ void norm_kernel(float* __restrict__ out,
                            const float* __restrict__ den12,
                            const float* __restrict__ den21,
                            int csplit, int Ctot, int HW, long N) {
  long idx = (long)blockIdx.x * blockDim.x + threadIdx.x;
  if (idx >= N) return;
  int pix = (int)(idx % HW);
  int c = (int)((idx / HW) % Ctot);
  int b = (int)(idx / ((long)HW * Ctot));
  const float* dp = (c < csplit) ? den12 : den21;
  float d = dp[(size_t)b * HW + pix];
  d = (d == 0.f) ? 1.f : d;
  out[idx] = out[idx] / d;
}

// ---------------- host orchestration -------------------------------------
extern "C" void kernel_launch(void* const* d_in, const int* in_sizes, int n_in,
                              void* d_out, int out_size, void* d_ws, size_t ws_size,
                              hipStream_t stream) {
  (void)in_sizes; (void)n_in; (void)out_size; (void)ws_size;
  const float* img1    = (const float*)d_in[0];
  const float* img2    = (const float*)d_in[1];
  const float* feat1_1 = (const float*)d_in[2];
  const float* feat1_2 = (const float*)d_in[3];
  const float* feat1_3 = (const float*)d_in[4];
  const float* feat2_1 = (const float*)d_in[5];
  const float* feat2_2 = (const float*)d_in[6];
  const float* feat2_3 = (const float*)d_in[7];
  const float* flow12  = (const float*)d_in[8];
  const float* flow21  = (const float*)d_in[9];
  const float* beta1   = (const float*)d_in[10];
  const float* beta2   = (const float*)d_in[11];

  float* out = (float*)d_out;
  float* ws  = (float*)d_ws;

  const int HW1 = H1_ * W1_, HW2 = H2_ * W2_, HW4 = H4_ * W4_;

  // 1) zero output + denominators (harness does not re-zero between replays)
  {
    long n4 = OUT_TOTAL / 4;
    zero4_kernel<<<(unsigned)((n4 + 255) / 256), 256, 0, stream>>>((float4*)out, n4);
    long d4 = DEN_LEN / 4;
    zero4_kernel<<<(unsigned)((d4 + 255) / 256), 256, 0, stream>>>((float4*)(ws + DEN_OFF), d4);
  }

  // 2) flow upsample (async-LDS staged)
  upflow_kernel<<<2 * B_ * 2 * H1_, W1_, 0, stream>>>(flow12, flow21, ws + RF_OFF, B_, H1_, W1_);
  upflow_kernel<<<2 * B_ * 2 * H2_, W2_, 0, stream>>>(flow12, flow21, ws + RH_OFF, B_, H2_, W2_);

  // 3) photometric metric at full res (both directions)
  metric_kernel<<<(2 * B_ * HW1 + 255) / 256, 256, 0, stream>>>(
      img1, img2, ws + RF_OFF, ws + M_OFF, B_, H1_, W1_);

  // 4) metric downsamples
  down_kernel<<<(2 * B_ * HW2 + 255) / 256, 256, 0, stream>>>(
      ws + M_OFF, ws + MH_OFF, 2 * B_, H1_, W1_, H2_, W2_, 0);
  down_kernel<<<(2 * B_ * HW4 + 255) / 256, 256, 0, stream>>>(
      ws + M_OFF, ws + MQ_OFF, 2 * B_, H1_, W1_, H4_, W4_, 1);

  // 5) forward splats (numerators -> d_out, denominators -> ws)
  const float* rF12 = ws + RF_OFF;
  const float* rF21 = ws + RF_OFF + (size_t)B_ * 2 * HW1;
  const float* rH12 = ws + RH_OFF;
  const float* rH21 = ws + RH_OFF + (size_t)B_ * 2 * HW2;
  const float* m12  = ws + M_OFF;
  const float* m21  = ws + M_OFF  + (size_t)B_ * HW1;
  const float* m12h = ws + MH_OFF;
  const float* m21h = ws + MH_OFF + (size_t)B_ * HW2;
  const float* m12q = ws + MQ_OFF;
  const float* m21q = ws + MQ_OFF + (size_t)B_ * HW4;

  int g1 = (B_ * HW1 + 255) / 256;
  int g2 = (B_ * HW2 + 255) / 256;
  int g3 = (B_ * HW4 + 255) / 256;

  // level 1 (full res, Ctot=70): [img1 | feat1_l1 | img2 | feat2_l1]
  splat_kernel<<<g1, 256, 0, stream>>>(img1,    3,  rF12, 10.0f, m12, beta1,
                                       out + OUT1_OFF, 0, 70, ws + DEN12_1, 1, B_, H1_, W1_);
  splat_kernel<<<g1, 256, 0, stream>>>(feat1_1, 32, rF12, 10.0f, m12, beta1,
                                       out + OUT1_OFF, 3, 70, ws + DEN12_1, 0, B_, H1_, W1_);
  splat_kernel<<<g1, 256, 0, stream>>>(img2,    3,  rF21, 10.0f, m21, beta2,
                                       out + OUT1_OFF, 35, 70, ws + DEN21_1, 1, B_, H1_, W1_);
  splat_kernel<<<g1, 256, 0, stream>>>(feat2_1, 32, rF21, 10.0f, m21, beta2,
                                       out + OUT1_OFF, 38, 70, ws + DEN21_1, 0, B_, H1_, W1_);

  // level 2 (half res, Ctot=128): [feat1_l2 | feat2_l2]
  splat_kernel<<<g2, 256, 0, stream>>>(feat1_2, 64, rH12, 5.0f, m12h, beta1,
                                       out + OUT2_OFF, 0, 128, ws + DEN12_2, 1, B_, H2_, W2_);
  splat_kernel<<<g2, 256, 0, stream>>>(feat2_2, 64, rH21, 5.0f, m21h, beta2,
                                       out + OUT2_OFF, 64, 128, ws + DEN21_2, 1, B_, H2_, W2_);

  // level 3 (quarter res, Ctot=192): [feat1_l3 | feat2_l3]; native flow * 2.5
  splat_kernel<<<g3, 256, 0, stream>>>(feat1_3, 96, flow12, 2.5f, m12q, beta1,
                                       out + OUT3_OFF, 0, 192, ws + DEN12_3, 1, B_, H4_, W4_);
  splat_kernel<<<g3, 256, 0, stream>>>(feat2_3, 96, flow21, 2.5f, m21q, beta2,
                                       out + OUT3_OFF, 96, 192, ws + DEN21_3, 1, B_, H4_, W4_);

  // 6) normalize each level in place
  {
    long n1 = (long)B_ * 70 * HW1;
    norm_kernel<<<(unsigned)((n1 + 255) / 256), 256, 0, stream>>>(
        out + OUT1_OFF, ws + DEN12_1, ws + DEN21_1, 35, 70, HW1, n1);
    long n2 = (long)B_ * 128 * HW2;
    norm_kernel<<<(unsigned)((n2 + 255) / 256), 256, 0, stream>>>(
        out + OUT2_OFF, ws + DEN12_2, ws + DEN21_2, 64, 128, HW2, n2);
    long n3 = (long)B_ * 192 * HW4;
    norm_kernel<<<(unsigned)((n3 + 255) / 256), 256, 0, stream>>>(
        out + OUT3_OFF, ws + DEN12_3, ws + DEN21_3, 96, 192, HW4, n3);
  }
}